// GATSingleAttentionHead_11828339933782
// MI455X (gfx1250) — compile-verified
//
#include <hip/hip_runtime.h>
#include <hip/hip_bf16.h>

#define N_NODES   50000
#define N_EDGES   1600000
#define IN_FEATS  256
#define OUT_FEATS 64

typedef __attribute__((ext_vector_type(2))) float v2f;
typedef __attribute__((ext_vector_type(8))) float v8f;
typedef __attribute__((ext_vector_type(4))) float v4f;

// ---------------------------------------------------------------------------
// helpers: monotone float<->uint encoding for atomic segment-max
// ---------------------------------------------------------------------------
__device__ __forceinline__ unsigned enc_f32(float f) {
    unsigned u = __float_as_uint(f);
    return (u & 0x80000000u) ? ~u : (u | 0x80000000u);
}
__device__ __forceinline__ float dec_f32(unsigned u) {
    unsigned b = (u & 0x80000000u) ? (u & 0x7FFFFFFFu) : ~u;
    return __uint_as_float(b);
}
__device__ __forceinline__ bool finite_f32(float f) {
    return (__float_as_uint(f) & 0x7F800000u) != 0x7F800000u;
}

// ---------------------------------------------------------------------------
// Kernel 1: Wh = feature @ W.T via V_WMMA_F32_16X16X4_F32.
// 8 waves / block, each wave computes a 16(M) x 64(N) strip.
// Also seeds d_out with Wh (the "+ Wh" residual term).
// ---------------------------------------------------------------------------
__global__ __launch_bounds__(256) void gat_gemm_wmma(
    const float* __restrict__ feature,   // [N_NODES, 256]
    const float* __restrict__ W,         // [64, 256] row-major == B(N,K) layout
    float* __restrict__ Wh,              // [N_NODES, 64]
    float* __restrict__ out)             // [N_NODES, 64] seeded = Wh
{
    const int wave = threadIdx.x >> 5;
    const int lane = threadIdx.x & 31;
    const int half = lane >> 4;      // 0: lanes 0-15, 1: lanes 16-31
    const int l16  = lane & 15;

    const int row_base = blockIdx.x * 128 + wave * 16;
    int m = row_base + l16;
    if (m >= N_NODES) m = N_NODES - 1;           // clamp: WMMA needs full EXEC

    const float* arow = feature + (size_t)m * IN_FEATS;

    v8f acc0 = {}, acc1 = {}, acc2 = {}, acc3 = {};

#pragma unroll 4
    for (int k0 = 0; k0 < IN_FEATS; k0 += 4) {
        // A fragment: lane holds A[M = l16][k0 + 2*half + {0,1}]
        v2f a = *(const v2f*)(arow + k0 + 2 * half);
        // B fragments: lane holds B[k][n] = W[n][k], n = t*16 + l16
        v2f b0 = *(const v2f*)(W + (size_t)(0 * 16 + l16) * IN_FEATS + k0 + 2 * half);
        v2f b1 = *(const v2f*)(W + (size_t)(1 * 16 + l16) * IN_FEATS + k0 + 2 * half);
        v2f b2 = *(const v2f*)(W + (size_t)(2 * 16 + l16) * IN_FEATS + k0 + 2 * half);
        v2f b3 = *(const v2f*)(W + (size_t)(3 * 16 + l16) * IN_FEATS + k0 + 2 * half);
        acc0 = __builtin_amdgcn_wmma_f32_16x16x4_f32(false, a, false, b0, (short)0, acc0, false, false);
        acc1 = __builtin_amdgcn_wmma_f32_16x16x4_f32(false, a, false, b1, (short)0, acc1, false, false);
        acc2 = __builtin_amdgcn_wmma_f32_16x16x4_f32(false, a, false, b2, (short)0, acc2, false, false);
        acc3 = __builtin_amdgcn_wmma_f32_16x16x4_f32(false, a, false, b3, (short)0, acc3, false, false);
    }

    // D layout: lane holds D[M = v + 8*half][N = l16] in VGPR v
#pragma unroll
    for (int v = 0; v < 8; ++v) {
        int mm = row_base + v + 8 * half;
        if (mm < N_NODES) {
            size_t base = (size_t)mm * OUT_FEATS + l16;
            float d0 = acc0[v], d1 = acc1[v], d2 = acc2[v], d3 = acc3[v];
            Wh[base + 0]  = d0;  out[base + 0]  = d0;
            Wh[base + 16] = d1;  out[base + 16] = d1;
            Wh[base + 32] = d2;  out[base + 32] = d2;
            Wh[base + 48] = d3;  out[base + 48] = d3;
        }
    }
}

// ---------------------------------------------------------------------------
// Kernel 2: per-node: el = Wh.a_src, er = Wh.a_dst; init max/denom arrays
// ---------------------------------------------------------------------------
__global__ void gat_node_prep(const float* __restrict__ Wh,
                              const float* __restrict__ att_w,   // [128]
                              float* __restrict__ el,
                              float* __restrict__ er,
                              unsigned* __restrict__ emax_enc,
                              float* __restrict__ denom)
{
    int n = blockIdx.x * blockDim.x + threadIdx.x;
    if (n >= N_NODES) return;
    const float* row = Wh + (size_t)n * OUT_FEATS;
    float accl = 0.f, accr = 0.f;
#pragma unroll
    for (int k = 0; k < OUT_FEATS; k += 4) {
        v4f h  = *(const v4f*)(row + k);
        v4f as = *(const v4f*)(att_w + k);
        v4f ad = *(const v4f*)(att_w + OUT_FEATS + k);
        accl += h.x * as.x + h.y * as.y + h.z * as.z + h.w * as.w;
        accr += h.x * ad.x + h.y * ad.y + h.z * ad.z + h.w * ad.w;
    }
    el[n] = accl;
    er[n] = accr;
    emax_enc[n] = 0u;   // decodes to NaN -> !finite -> 0 (matches -inf path)
    denom[n] = 0.f;
}

// ---------------------------------------------------------------------------
// Kernel 3: per-edge logits + atomic segment-max over dst
// ---------------------------------------------------------------------------
__global__ void gat_edge_logits(const int* __restrict__ src,
                                const int* __restrict__ dst,
                                const float* __restrict__ el,
                                const float* __restrict__ er,
                                float* __restrict__ e_buf,
                                unsigned* __restrict__ emax_enc)
{
    int i = blockIdx.x * blockDim.x + threadIdx.x;
    if (i >= N_EDGES) return;
    float e = el[src[i]] + er[dst[i]];
    e = (e > 0.f) ? e : 0.2f * e;          // leaky_relu, slope 0.2
    e_buf[i] = e;
    atomicMax(&emax_enc[dst[i]], enc_f32(e));
}

// ---------------------------------------------------------------------------
// Kernel 4: decode per-node max; non-finite (incl. empty segment) -> 0
// ---------------------------------------------------------------------------
__global__ void gat_node_max(const unsigned* __restrict__ emax_enc,
                             float* __restrict__ emax_f)
{
    int n = blockIdx.x * blockDim.x + threadIdx.x;
    if (n >= N_NODES) return;
    float f = dec_f32(emax_enc[n]);
    emax_f[n] = finite_f32(f) ? f : 0.f;
}

// ---------------------------------------------------------------------------
// Kernel 5: per-edge exp + atomic segment-sum of denom
// ---------------------------------------------------------------------------
__global__ void gat_edge_exp(const int* __restrict__ dst,
                             const float* __restrict__ emax_f,
                             float* __restrict__ e_buf,     // in: e, out: exp
                             float* __restrict__ denom)
{
    int i = blockIdx.x * blockDim.x + threadIdx.x;
    if (i >= N_EDGES) return;
    int d = dst[i];
    float ex = expf(e_buf[i] - emax_f[d]);
    e_buf[i] = ex;
    atomicAdd(&denom[d], ex);
}

// ---------------------------------------------------------------------------
// Kernel 6: per-node reciprocal of denom (denom<=0 -> divide by 1)
// ---------------------------------------------------------------------------
__global__ void gat_node_inv(float* __restrict__ denom)
{
    int n = blockIdx.x * blockDim.x + threadIdx.x;
    if (n >= N_NODES) return;
    float d = denom[n];
    denom[n] = (d > 0.f) ? (1.0f / d) : 1.0f;
}

// ---------------------------------------------------------------------------
// Kernel 7: scatter-aggregate: out[dst] += alpha * Wh[src]
// 64 threads per edge (one per output feature); atomics stay in L2.
// ---------------------------------------------------------------------------
__global__ __launch_bounds__(256) void gat_edge_aggregate(
    const int* __restrict__ src,
    const int* __restrict__ dst,
    const float* __restrict__ e_exp,
    const float* __restrict__ inv_denom,
    const float* __restrict__ Wh,
    float* __restrict__ out)
{
    int t = blockIdx.x * blockDim.x + threadIdx.x;
    int edge = t >> 6;              // / OUT_FEATS
    int f    = t & 63;
    if (edge >= N_EDGES) return;
    int s = src[edge];
    int d = dst[edge];
    float alpha = e_exp[edge] * inv_denom[d];
    float val = alpha * Wh[(size_t)s * OUT_FEATS + f];
    atomicAdd(&out[(size_t)d * OUT_FEATS + f], val);
}

// ---------------------------------------------------------------------------
// Kernel 8: finalize: out = elu(out + bias)
// ---------------------------------------------------------------------------
__global__ void gat_node_final(float* __restrict__ out,
                               const float* __restrict__ bias)  // [64]
{
    int i = blockIdx.x * blockDim.x + threadIdx.x;
    if (i >= N_NODES * OUT_FEATS) return;
    float x = out[i] + bias[i & 63];
    out[i] = (x > 0.f) ? x : (expf(x) - 1.0f);
}

// ---------------------------------------------------------------------------
// launch
// ---------------------------------------------------------------------------
extern "C" void kernel_launch(void* const* d_in, const int* in_sizes, int n_in,
                              void* d_out, int out_size, void* d_ws, size_t ws_size,
                              hipStream_t stream)
{
    const float* feature = (const float*)d_in[0];   // [50000,256]
    const float* W       = (const float*)d_in[1];   // [64,256]
    const float* att_w   = (const float*)d_in[2];   // [1,128]
    const float* bias    = (const float*)d_in[3];   // [1,64]
    const int*   src     = (const int*)d_in[4];     // [1.6M]
    const int*   dst     = (const int*)d_in[5];     // [1.6M]
    float*       out     = (float*)d_out;           // [50000,64]

    // workspace layout (floats)
    float* ws = (float*)d_ws;
    float*    Wh        = ws;                               // 3,200,000
    float*    el        = Wh + (size_t)N_NODES * OUT_FEATS; // 50,000
    float*    er        = el + N_NODES;                     // 50,000
    unsigned* emax_enc  = (unsigned*)(er + N_NODES);        // 50,000
    float*    denom     = (float*)(emax_enc + N_NODES);     // 50,000 (-> inv)
    float*    emax_f    = denom + N_NODES;                  // 50,000
    float*    e_buf     = emax_f + N_NODES;                 // 1,600,000 (-> exp)

    const int nodeBlocks = (N_NODES + 255) / 256;
    const int edgeBlocks = (N_EDGES + 255) / 256;

    // 1. GEMM (128 rows per 256-thread block)
    gat_gemm_wmma<<<(N_NODES + 127) / 128, 256, 0, stream>>>(feature, W, Wh, out);
    // 2. node prep
    gat_node_prep<<<nodeBlocks, 256, 0, stream>>>(Wh, att_w, el, er, emax_enc, denom);
    // 3. edge logits + segment max
    gat_edge_logits<<<edgeBlocks, 256, 0, stream>>>(src, dst, el, er, e_buf, emax_enc);
    // 4. decode max
    gat_node_max<<<nodeBlocks, 256, 0, stream>>>(emax_enc, emax_f);
    // 5. exp + segment sum
    gat_edge_exp<<<edgeBlocks, 256, 0, stream>>>(dst, emax_f, e_buf, denom);
    // 6. reciprocal
    gat_node_inv<<<nodeBlocks, 256, 0, stream>>>(denom);
    // 7. scatter aggregate (64 threads per edge)
    {
        long long total = (long long)N_EDGES * OUT_FEATS;
        int blocks = (int)((total + 255) / 256);
        gat_edge_aggregate<<<blocks, 256, 0, stream>>>(src, dst, e_buf, denom, Wh, out);
    }
    // 8. bias + ELU
    gat_node_final<<<(N_NODES * OUT_FEATS + 255) / 256, 256, 0, stream>>>(out, bias);
}